// MultiGRU_21028159881529
// MI455X (gfx1250) — compile-verified
//
#include <hip/hip_runtime.h>
#include <math.h>

typedef __attribute__((ext_vector_type(2))) float v2f;
typedef __attribute__((ext_vector_type(8))) float v8f;

#define NHEAD 24
#define HID   12
#define IN_DIM 256
#define CAT   268           // HID + IN_DIM
#define BATCH 8192
#define WAVES_PER_WG 4
#define ROWS_PER_WAVE 16
#define WPITCH 16           // weights padded out-dim (12 -> 16, cols 12..15 = 0)
#define RHPITCH 20          // r*h tile pitch chosen for conflict-free A-frag reads

__global__ __launch_bounds__(WAVES_PER_WG * 32)
void multigru_wmma_f32_kernel(const float* __restrict__ h,
                              const float* __restrict__ x,
                              const float* __restrict__ Wz, const float* __restrict__ bz,
                              const float* __restrict__ Wr, const float* __restrict__ br,
                              const float* __restrict__ Wq, const float* __restrict__ bq,
                              float* __restrict__ out)
{
    // 3 * 268 * 16 * 4B = 51456 B of padded weights, + 4 * 16 * 20 * 4B r*h tiles
    __shared__ float lw[3][CAT][WPITCH];
    __shared__ float lrh[WAVES_PER_WG][ROWS_PER_WAVE][RHPITCH];

    const int tid  = threadIdx.x;
    const int wave = tid >> 5;
    const int lane = tid & 31;
    const int head = blockIdx.y;
    const int b0   = (blockIdx.x * WAVES_PER_WG + wave) * ROWS_PER_WAVE;

    // ---- cooperative stage of Wz/Wr/Wq into LDS, zero-padded to 16 cols ----
    for (int mat = 0; mat < 3; ++mat) {
        const float* Wsrc = (mat == 0) ? Wz : (mat == 1) ? Wr : Wq;
        for (int idx = tid; idx < CAT * WPITCH; idx += WAVES_PER_WG * 32) {
            const int k = idx >> 4;
            const int n = idx & 15;
            lw[mat][k][n] = (n < HID) ? Wsrc[(head * CAT + k) * HID + n] : 0.0f;
        }
    }
    __syncthreads();

    // ---- per-lane WMMA coordinates (ISA 7.12.2 layouts, wave32) ----
    const int half = lane >> 4;       // 0: K pair {0,1} / rows M=0..7 ; 1: {2,3} / M=8..15
    const int mrow = lane & 15;       // A-frag row
    const int ncol = lane & 15;       // B/C/D column
    const int kb   = half * 2;        // K sub-offset within a 4-wide step

    v8f accZ = {};
    v8f accR = {};
    v8f accQ = {};

    const float* hrow = h + (size_t)(b0 + mrow) * (NHEAD * HID)    + head * HID;
    const float* xrow = x + (size_t)(b0 + mrow) * (NHEAD * IN_DIM) + head * IN_DIM;

    // ---- K = 0..11 : h-part of hx -> z and r only (q's h-part needs r first) ----
#pragma unroll
    for (int k0 = 0; k0 < HID; k0 += 4) {
        v2f a = *(const v2f*)(hrow + k0 + kb);
        v2f bz2, br2;
        bz2.x = lw[0][k0 + kb    ][ncol];
        bz2.y = lw[0][k0 + kb + 1][ncol];
        br2.x = lw[1][k0 + kb    ][ncol];
        br2.y = lw[1][k0 + kb + 1][ncol];
        accZ = __builtin_amdgcn_wmma_f32_16x16x4_f32(false, a, false, bz2, (short)0, accZ, false, false);
        accR = __builtin_amdgcn_wmma_f32_16x16x4_f32(false, a, false, br2, (short)0, accR, false, false);
    }

    // ---- K = 12..267 : x-part of hx -> z, r and q (x-part identical for rhx) ----
#pragma unroll 4
    for (int k0 = HID; k0 < CAT; k0 += 4) {
        v2f a = *(const v2f*)(xrow + (k0 - HID) + kb);
        v2f bz2, br2, bq2;
        bz2.x = lw[0][k0 + kb    ][ncol];
        bz2.y = lw[0][k0 + kb + 1][ncol];
        br2.x = lw[1][k0 + kb    ][ncol];
        br2.y = lw[1][k0 + kb + 1][ncol];
        bq2.x = lw[2][k0 + kb    ][ncol];
        bq2.y = lw[2][k0 + kb + 1][ncol];
        accZ = __builtin_amdgcn_wmma_f32_16x16x4_f32(false, a, false, bz2, (short)0, accZ, false, false);
        accR = __builtin_amdgcn_wmma_f32_16x16x4_f32(false, a, false, br2, (short)0, accR, false, false);
        accQ = __builtin_amdgcn_wmma_f32_16x16x4_f32(false, a, false, bq2, (short)0, accQ, false, false);
    }

    // ---- activations for z, r ; build r*h tile in LDS (D-layout -> row-major) ----
    const float biasz = (ncol < HID) ? bz[head * HID + ncol] : 0.0f;
    const float biasr = (ncol < HID) ? br[head * HID + ncol] : 0.0f;
    const float biasq = (ncol < HID) ? bq[head * HID + ncol] : 0.0f;

    float hv[8], zv[8];
#pragma unroll
    for (int j = 0; j < 8; ++j) {
        const int m = j + half * 8;   // output row index in D layout
        hv[j] = (ncol < HID) ? h[(size_t)(b0 + m) * (NHEAD * HID) + head * HID + ncol] : 0.0f;
        zv[j] = 1.0f / (1.0f + __expf(-(accZ[j] + biasz)));
        const float rj = 1.0f / (1.0f + __expf(-(accR[j] + biasr)));
        lrh[wave][m][ncol] = rj * hv[j];          // cols >= HID: hv==0 -> 0 padding
    }
    __syncthreads();   // order LDS writes before A-frag reads (uniform control flow)

    // ---- K = 0..11 of rhx : (r*h) @ Wq[0:12,:] ----
#pragma unroll
    for (int k0 = 0; k0 < HID; k0 += 4) {
        v2f a, bq2;
        a.x   = lrh[wave][mrow][k0 + kb];
        a.y   = lrh[wave][mrow][k0 + kb + 1];
        bq2.x = lw[2][k0 + kb    ][ncol];
        bq2.y = lw[2][k0 + kb + 1][ncol];
        accQ = __builtin_amdgcn_wmma_f32_16x16x4_f32(false, a, false, bq2, (short)0, accQ, false, false);
    }

    // ---- epilogue: out = (1-z)*h + z*tanh(q) ----
#pragma unroll
    for (int j = 0; j < 8; ++j) {
        const int m = j + half * 8;
        const float q   = tanhf(accQ[j] + biasq);
        const float res = (1.0f - zv[j]) * hv[j] + zv[j] * q;
        if (ncol < HID)
            out[(size_t)(b0 + m) * (NHEAD * HID) + head * HID + ncol] = res;
    }
}

extern "C" void kernel_launch(void* const* d_in, const int* in_sizes, int n_in,
                              void* d_out, int out_size, void* d_ws, size_t ws_size,
                              hipStream_t stream) {
    (void)in_sizes; (void)n_in; (void)out_size; (void)d_ws; (void)ws_size;
    const float* h  = (const float*)d_in[0];
    const float* x  = (const float*)d_in[1];
    const float* Wz = (const float*)d_in[2];
    const float* bz = (const float*)d_in[3];
    const float* Wr = (const float*)d_in[4];
    const float* br = (const float*)d_in[5];
    const float* Wq = (const float*)d_in[6];
    const float* bq = (const float*)d_in[7];
    float* out = (float*)d_out;

    dim3 grid(BATCH / (WAVES_PER_WG * ROWS_PER_WAVE), NHEAD);   // (128, 24)
    multigru_wmma_f32_kernel<<<grid, WAVES_PER_WG * 32, 0, stream>>>(
        h, x, Wz, bz, Wr, br, Wq, bq, out);
}